// NFFTBase_46926812676636
// MI455X (gfx1250) — compile-verified
//
#include <hip/hip_runtime.h>
#include <math.h>

// ---------------------------------------------------------------------------
// KB-NUFFT forward (apodize -> pad/ifftshift -> ortho FFT2 -> KB interp)
// MI455X / gfx1250. FFT stages on the matrix pipe; LDS staging via CDNA5
// async global->LDS copies (ASYNCcnt) with interleaved complex LDS layout.
//   FFT 944  = 16 x 59: radix-16 WMMA + DFT-59 WMMA (K pad 59->60, M 59->64)
//   FFT 1280 = 16 x 80: radix-16 WMMA + DFT-80 WMMA (exact tiling)
// Intermediates in d_ws (2 grids x 38.7 MB, L2-resident on 192 MB L2).
// ---------------------------------------------------------------------------

typedef __attribute__((ext_vector_type(2))) float v2f;
typedef __attribute__((ext_vector_type(8))) float v8f;

#define JKB   6
#define IM0   640
#define IM1   472
#define G0    1280
#define G1    944
#define BATCH 4
#define NK    100000
#define PI_F  3.14159265358979f
#define BETA_F 13.8551006f   // pi*sqrt((J/2)^2*(1.5)^2 - 0.8) = pi*sqrt(19.45)

#define WMMA4(A, B, C) __builtin_amdgcn_wmma_f32_16x16x4_f32(false, A, false, B, (short)0, C, false, false)

__device__ __forceinline__ unsigned lds_addr_of(const void* p) {
    return (unsigned)(unsigned long long)p;   // low 32 bits of generic ptr = LDS offset
}
__device__ __forceinline__ void async_ldsload_b128(unsigned lds, const void* gaddr) {
    asm volatile("global_load_async_to_lds_b128 %0, %1, off"
                 :: "v"(lds), "v"((unsigned long long)gaddr) : "memory");
}
__device__ __forceinline__ void async_ldsload_b64(unsigned lds, const void* gaddr) {
    asm volatile("global_load_async_to_lds_b64 %0, %1, off"
                 :: "v"(lds), "v"((unsigned long long)gaddr) : "memory");
}
__device__ __forceinline__ void wait_async0() {
    asm volatile("s_wait_asynccnt 0" ::: "memory");
}

// ---------------- Kaiser-Bessel I0 (Abramowitz & Stegun 9.8.1/9.8.2) -------
__device__ __forceinline__ float bessel_i0(float x) {
    float ax = fabsf(x);
    if (ax < 3.75f) {
        float t = x / 3.75f; t *= t;
        return 1.0f + t*(3.5156229f + t*(3.0899424f + t*(1.2067492f +
                     t*(0.2659732f + t*(0.0360768f + t*0.0045813f)))));
    }
    float t = 3.75f / ax;
    return (expf(ax) * __frsqrt_rn(ax)) *
           (0.39894228f + t*(0.01328592f + t*(0.00225319f + t*(-0.00157565f +
            t*(0.00916281f + t*(-0.02057706f + t*(0.02635537f +
            t*(-0.01647633f + t*0.00392377f))))))));
}

// ---------------- Kernel 1: apodize + centered pad + ifftshift -------------
__global__ void k_pad(const float* __restrict__ ir, const float* __restrict__ ii,
                      float2* __restrict__ g) {
    long e = (long)blockIdx.x * blockDim.x + threadIdx.x;
    const long total = (long)BATCH * G0 * G1;
    if (e >= total) return;
    int c = (int)(e % G1);
    long t = e / G1;
    int r = (int)(t % G0);
    int b = (int)(t / G0);
    int mr = r + G0/2; if (mr >= G0) mr -= G0;   // ifftshift
    int mc = c + G1/2; if (mc >= G1) mc -= G1;
    int i = mr - (G0 - IM0)/2;
    int j = mc - (G1 - IM1)/2;
    float2 v = make_float2(0.f, 0.f);
    if (i >= 0 && i < IM0 && j >= 0 && j < IM1) {
        float re = ir[((long)b*IM0 + i)*IM1 + j];
        float im = ii[((long)b*IM0 + i)*IM1 + j];
        float x0 = (float)(i - IM0/2) / (float)G0;
        float z0 = BETA_F*BETA_F - (PI_F*JKB*x0)*(PI_F*JKB*x0);
        float s0 = sqrtf(fmaxf(z0, 1e-12f));
        float a0 = (float)JKB * sinhf(s0) / s0;
        float x1 = (float)(j - IM1/2) / (float)G1;
        float z1 = BETA_F*BETA_F - (PI_F*JKB*x1)*(PI_F*JKB*x1);
        float s1 = sqrtf(fmaxf(z1, 1e-12f));
        float a1 = (float)JKB * sinhf(s1) / s1;
        float inv = 1.0f / (a0 * a1);
        v = make_float2(re*inv, im*inv);
    }
    g[e] = v;
}

// ---------------- Radix-16 stage via WMMA f32 16x16x4 ----------------------
// X: interleaved complex LDS, slot = row*rowlenS + N2S*n1 + n2.
// A-matrix (16x4): lane<16 -> M=lane, K={k0,k0+1}; lane>=16 -> K={k0+2,k0+3}
// B-matrix (4x16) mirrored in K; D (16x16): VGPR j -> M=j+8*half, N=lane&15.
__device__ __forceinline__ void radix16_wmma(float2* X,
                                             int N2, int N2S, int rowlenS,
                                             int ntiles, int wave, int nwaves,
                                             int lane) {
    const int half = lane >> 4;
    const int lm   = lane & 15;
    const float w16 = -2.0f * PI_F / 16.0f;
    v2f aC[4], aS[4], aSn[4];
#pragma unroll
    for (int ch = 0; ch < 4; ++ch) {
        int kA = 4*ch + 2*half;
        float s0, c0, s1, c1;
        __sincosf(w16 * (float)(lm * kA),       &s0, &c0);
        __sincosf(w16 * (float)(lm * (kA + 1)), &s1, &c1);
        aC[ch].x = c0;   aC[ch].y = c1;
        aS[ch].x = s0;   aS[ch].y = s1;
        aSn[ch].x = -s0; aSn[ch].y = -s1;
    }
    for (int t = wave; t < ntiles; t += nwaves) {
        int col  = 16*t + lm;          // flattened (row, n2)
        int row  = col / N2;
        int n2   = col % N2;
        int base = row*rowlenS + n2;
        v2f bR[4], bI[4];
#pragma unroll
        for (int ch = 0; ch < 4; ++ch) {
            int kA = 4*ch + 2*half;
            float2 f0 = X[base + N2S*kA];
            float2 f1 = X[base + N2S*(kA+1)];
            bR[ch].x = f0.x; bR[ch].y = f1.x;
            bI[ch].x = f0.y; bI[ch].y = f1.y;
        }
        v8f cr = {0.f,0.f,0.f,0.f,0.f,0.f,0.f,0.f};
        v8f ci = {0.f,0.f,0.f,0.f,0.f,0.f,0.f,0.f};
#pragma unroll
        for (int ch = 0; ch < 4; ++ch) cr = WMMA4(aC[ch],  bR[ch], cr);
#pragma unroll
        for (int ch = 0; ch < 4; ++ch) cr = WMMA4(aSn[ch], bI[ch], cr);
#pragma unroll
        for (int ch = 0; ch < 4; ++ch) ci = WMMA4(aC[ch],  bI[ch], ci);
#pragma unroll
        for (int ch = 0; ch < 4; ++ch) ci = WMMA4(aS[ch],  bR[ch], ci);
#pragma unroll
        for (int jj = 0; jj < 8; ++jj) {
            int m = jj + 8*half;
            X[row*rowlenS + N2S*m + n2] = make_float2(cr[jj], ci[jj]);
        }
    }
}

// ---------------- Stage-B tile: D(16x16) = W(16xK) * Z(Kx16), complex ------
__device__ __forceinline__ void dftN_wmma_tile(const float2* __restrict__ X,
                                               const float2* __restrict__ W,
                                               int Wstride, int mt, int nt,
                                               int rowlenS, int N2S, int nchunks,
                                               int lane, v8f& cr, v8f& ci) {
    const int half = lane >> 4;
    const int lm   = lane & 15;
    const int col  = 16*nt + lm;
    const int row  = col >> 4;
    const int k1   = col & 15;
    const int zb   = row*rowlenS + N2S*k1;
    const int wb   = (16*mt + lm)*Wstride;
    cr = (v8f){0.f,0.f,0.f,0.f,0.f,0.f,0.f,0.f};
    ci = (v8f){0.f,0.f,0.f,0.f,0.f,0.f,0.f,0.f};
    for (int ch = 0; ch < nchunks; ++ch) {
        int kA = 4*ch + 2*half;
        float2 w0 = W[wb + kA];
        float2 w1 = W[wb + kA + 1];
        float2 z0 = X[zb + kA];      // contiguous 16B pair with z1
        float2 z1 = X[zb + kA + 1];
        v2f aC;  aC.x  = w0.x;  aC.y  = w1.x;
        v2f aS;  aS.x  = w0.y;  aS.y  = w1.y;
        v2f aSn; aSn.x = -w0.y; aSn.y = -w1.y;
        v2f bR;  bR.x  = z0.x;  bR.y  = z1.x;
        v2f bI;  bI.x  = z0.y;  bI.y  = z1.y;
        cr = WMMA4(aC,  bR, cr);
        cr = WMMA4(aSn, bI, cr);
        ci = WMMA4(aC,  bI, ci);
        ci = WMMA4(aS,  bR, ci);
    }
}

// ---------------- Kernel 2: length-944 FFT over rows, transposed out -------
// Padded n1-stride 60; staged via per-element async B64 copies (layout map).
__global__ __launch_bounds__(256) void k_fft944(const float2* __restrict__ gin,
                                                float2* __restrict__ gt) {
    __shared__ __align__(16) float2 X[16 * 960];
    __shared__ __align__(16) float2 W59[64 * 60];   // zero-padded 59x59 table
    const int tid  = threadIdx.x;
    const int wave = tid >> 5, lane = tid & 31;
    const int b  = blockIdx.x / (G0/16);
    const int r0 = (blockIdx.x % (G0/16)) * 16;

    // async global->LDS stage of 16 rows (padded slots), overlapped with
    // the W59 table build below.
    const float2* src = &gin[(long)(b*G0 + r0) * G1];
    for (int e = tid; e < 16*G1; e += 256) {
        int row = e / G1, n = e % G1;
        int n1 = n / 59, n2 = n - 59*n1;
        int slot = row*960 + 60*n1 + n2;
        async_ldsload_b64(lds_addr_of(&X[slot]), src + e);
    }
    for (int e = tid; e < 16*16; e += 256) {   // zero the K-pad column n2=59
        X[(e >> 4)*960 + 60*(e & 15) + 59] = make_float2(0.f, 0.f);
    }
    for (int e = tid; e < 64*60; e += 256) {
        int k2 = e / 60, n2 = e % 60;
        float2 w = make_float2(0.f, 0.f);
        if (k2 < 59 && n2 < 59) {
            float s, c; __sincosf(-2.0f*PI_F*(float)(k2*n2)/59.0f, &s, &c);
            w = make_float2(c, s);
        }
        W59[e] = w;
    }
    wait_async0();
    __syncthreads();
    radix16_wmma(X, 59, 60, 960, 59, wave, 8, lane);
    __syncthreads();
    // twiddle W_944^{k1*n2} (pad slots hold zero; multiply is harmless)
    for (int e = tid; e < 16*960; e += 256) {
        int rem = e % 960;
        int k1 = rem / 60, n2 = rem - 60*k1;
        float s, c; __sincosf(-2.0f*PI_F*(float)(k1*n2)/(float)G1, &s, &c);
        float2 v = X[e];
        X[e] = make_float2(v.x*c - v.y*s, v.x*s + v.y*c);
    }
    __syncthreads();
    // stage B: DFT-59 via WMMA. Tiles: 4 (k2, padded to 64) x 16 (row,k1).
    const int half = lane >> 4, lm = lane & 15;
    for (int tt = wave; tt < 64; tt += 8) {
        int mt = tt >> 4, nt = tt & 15;
        v8f cr, ci;
        dftN_wmma_tile(X, W59, 60, mt, nt, 960, 60, 15, lane, cr, ci);
        int col = 16*nt + lm;
        int row = col >> 4, k1 = col & 15;
#pragma unroll
        for (int jj = 0; jj < 8; ++jj) {
            int k2 = 16*mt + jj + 8*half;
            if (k2 < 59) {
                int nout = k1 + 16*k2;
                gt[((long)(b*G1 + nout))*G0 + r0 + row] = make_float2(cr[jj], ci[jj]);
            }
        }
    }
}

// ---------------- Kernel 3: length-1280 FFT over transposed rows -----------
// 16 input rows are one contiguous 160 KB block: async B128 stage.
__global__ __launch_bounds__(256) void k_fft1280(const float2* __restrict__ gt,
                                                 float2* __restrict__ gout) {
    __shared__ __align__(16) float2 X[16 * G0];
    __shared__ __align__(16) float2 W80[80 * 80];
    const int tid  = threadIdx.x;
    const int wave = tid >> 5, lane = tid & 31;
    const int b  = blockIdx.x / (G1/16);
    const int w0 = (blockIdx.x % (G1/16)) * 16;

    const float2* src = &gt[(long)(b*G1 + w0) * G0];
    for (int q = tid; q < (16*G0)/2; q += 256) {        // 16B chunks
        async_ldsload_b128(lds_addr_of(&X[2*q]), src + 2*q);
    }
    for (int e = tid; e < 80*80; e += 256) {
        int k2 = e / 80, n2 = e % 80;
        float s, c; __sincosf(-2.0f*PI_F*(float)(k2*n2)/80.0f, &s, &c);
        W80[e] = make_float2(c, s);
    }
    wait_async0();
    __syncthreads();
    radix16_wmma(X, 80, 80, G0, 80, wave, 8, lane);
    __syncthreads();
    for (int e = tid; e < 16*G0; e += 256) {
        int rem = e % G0;
        int k1 = rem / 80, n2 = rem - 80*k1;
        float s, c; __sincosf(-2.0f*PI_F*(float)(k1*n2)/(float)G0, &s, &c);
        float2 v = X[e];
        X[e] = make_float2(v.x*c - v.y*s, v.x*s + v.y*c);
    }
    __syncthreads();
    // stage B: DFT-80 via WMMA. Tiles: 5 (k2) x 16 (row,k1); K = 80 exact.
    const float scale = 1.0f / sqrtf((float)(G0) * (float)(G1)); // ortho norm
    const int half = lane >> 4, lm = lane & 15;
    for (int tt = wave; tt < 80; tt += 8) {
        int mt = tt / 16, nt = tt & 15;
        v8f cr, ci;
        dftN_wmma_tile(X, W80, 80, mt, nt, G0, 80, 20, lane, cr, ci);
        int col = 16*nt + lm;
        int row = col >> 4, k1 = col & 15;
#pragma unroll
        for (int jj = 0; jj < 8; ++jj) {
            int k2 = 16*mt + jj + 8*half;
            int kh = k1 + 16*k2;
            gout[((long)(b*G0 + kh))*G1 + w0 + row] =
                make_float2(cr[jj]*scale, ci[jj]*scale);
        }
    }
}

// ---------------- Kernel 4: 6x6 Kaiser-Bessel interpolation ----------------
__global__ void k_interp(const float* __restrict__ kt, const float2* __restrict__ gk,
                         float2* __restrict__ out) {
    int e = blockIdx.x * blockDim.x + threadIdx.x;
    if (e >= BATCH * NK) return;
    int b = e / NK, k = e % NK;
    float t0 = kt[((long)b*2 + 0)*NK + k] * ((float)G0 / (2.0f*PI_F));
    float t1 = kt[((long)b*2 + 1)*NK + k] * ((float)G1 / (2.0f*PI_F));
    int b0 = (int)floorf(t0);
    int b1 = (int)floorf(t1);
    float wx[JKB], wy[JKB];
    int   ix[JKB], iy[JKB];
#pragma unroll
    for (int jj = 0; jj < JKB; ++jj) {
        int kidx = b0 + jj - (JKB/2 - 1);
        float u = t0 - (float)kidx;
        float z = 1.0f - (u/3.0f)*(u/3.0f);
        wx[jj] = (z > 0.f) ? bessel_i0(BETA_F * sqrtf(z)) : 0.f;
        int m = kidx % G0; if (m < 0) m += G0;
        ix[jj] = m;
    }
#pragma unroll
    for (int jj = 0; jj < JKB; ++jj) {
        int kidx = b1 + jj - (JKB/2 - 1);
        float u = t1 - (float)kidx;
        float z = 1.0f - (u/3.0f)*(u/3.0f);
        wy[jj] = (z > 0.f) ? bessel_i0(BETA_F * sqrtf(z)) : 0.f;
        int m = kidx % G1; if (m < 0) m += G1;
        iy[jj] = m;
    }
    float ar = 0.f, ai = 0.f;
#pragma unroll
    for (int jx = 0; jx < JKB; ++jx) {
        const float2* rowp = &gk[((long)b*G0 + ix[jx])*G1];
        float wxx = wx[jx];
#pragma unroll
        for (int jy = 0; jy < JKB; ++jy) {
            float2 v = rowp[iy[jy]];
            float w = wxx * wy[jy];
            ar += w * v.x;
            ai += w * v.y;
        }
    }
    out[e] = make_float2(ar, ai);
}

// ---------------------------------------------------------------------------
extern "C" void kernel_launch(void* const* d_in, const int* in_sizes, int n_in,
                              void* d_out, int out_size, void* d_ws, size_t ws_size,
                              hipStream_t stream) {
    (void)in_sizes; (void)n_in; (void)out_size; (void)ws_size;
    const float* image_r = (const float*)d_in[0];
    const float* image_i = (const float*)d_in[1];
    const float* ktraj   = (const float*)d_in[2];
    float2* gridA = (float2*)d_ws;                          // [B][1280][944]
    float2* gridT = gridA + (size_t)BATCH * G0 * G1;        // [B][944][1280]
    float2* out   = (float2*)d_out;                         // [B][K] complex

    const long total = (long)BATCH * G0 * G1;
    k_pad    <<<dim3((unsigned)((total + 255)/256)), dim3(256), 0, stream>>>(image_r, image_i, gridA);
    k_fft944 <<<dim3(BATCH*(G0/16)), dim3(256), 0, stream>>>(gridA, gridT);
    k_fft1280<<<dim3(BATCH*(G1/16)), dim3(256), 0, stream>>>(gridT, gridA);
    k_interp <<<dim3((BATCH*NK + 255)/256), dim3(256), 0, stream>>>(ktraj, gridA, out);
}